// LRU_48438641164820
// MI455X (gfx1250) — compile-verified
//
#include <hip/hip_runtime.h>
#include <math.h>

typedef __attribute__((ext_vector_type(2))) float v2f;
typedef __attribute__((ext_vector_type(8))) float v8f;

#define BSZ   2
#define T_LEN 2048
#define D_DIM 128
#define S_DIM 64
#define CH    16   // time-chunk length == WMMA M

// DPP row-reduction step: add value shifted from lower lanes within the
// 16-lane row (bound_ctrl -> zero fill). After ctrls 0x111,0x112,0x114,0x118
// lane 15 of each row holds the sum of its 16-lane row. Pure VALU, no LDS.
template <int CTRL>
__device__ __forceinline__ float dpp_row_add(float v) {
    int sh = __builtin_amdgcn_update_dpp(0, __builtin_bit_cast(int, v),
                                         CTRL, 0xF, 0xF, true);
    return v + __builtin_bit_cast(float, sh);
}

// One workgroup per (b, d); 4 waves, wave w owns states s in [16w, 16w+16).
// Intra-chunk inclusive scan via cs = L * s with L = lower-triangular ones,
// as 4 chained V_WMMA_F32_16X16X4_F32 per (real, imag). All decay powers are
// folded into loop-invariant per-lane coefficients. x staging uses the CDNA5
// async global->LDS engine (ASYNCcnt) instead of a VGPR round-trip.
__global__ __launch_bounds__(128) void lru_ssd_wmma_kernel(
    const float* __restrict__ x,
    const float* __restrict__ log_a_mag,
    const float* __restrict__ a_phase,
    const float* __restrict__ B_re, const float* __restrict__ B_im,
    const float* __restrict__ C_re, const float* __restrict__ C_im,
    const float* __restrict__ D_diag,
    float* __restrict__ y)
{
    __shared__ float xbuf[CH];
    __shared__ float ybuf[4][CH];
    __shared__ float Hr[S_DIM], Hi[S_DIM];

    const int tid  = threadIdx.x;
    const int wave = tid >> 5;
    const int lane = tid & 31;
    const int sl   = lane & 15;   // state-local index == WMMA N / A's M row
    const int hh   = lane >> 4;   // half-wave select
    const int bid  = blockIdx.x;
    const int b    = bid / D_DIM;
    const int d    = bid % D_DIM;
    const int s    = wave * 16 + sl;

    const float lm = log_a_mag[d * S_DIM + s];
    const float ph = a_phase [d * S_DIM + s];
    const float Br = B_re    [d * S_DIM + s];
    const float Bi = B_im    [d * S_DIM + s];
    const float Cr = C_re    [d * S_DIM + s];
    const float Ci = C_im    [d * S_DIM + s];
    const float Dd = D_diag  [d];

    // a^1 (per state)
    float sn1, cs1; __sincosf(ph, &sn1, &cs1);
    const float e1  = __expf(lm);
    const float a1r = e1 * cs1, a1i = e1 * sn1;

    // A operand: lower-triangular ones, 16x16 split into 4 K-slices of 16x4.
    // A layout (32-bit 16x4): lane M = lane&15; reg r holds K = 2*(lane>>4)+r.
    v2f Amat[4];
    #pragma unroll
    for (int q = 0; q < 4; ++q) {
        Amat[q].x = ((4 * q + 2 * hh + 0) <= sl) ? 1.0f : 0.0f;
        Amat[q].y = ((4 * q + 2 * hh + 1) <= sl) ? 1.0f : 0.0f;
    }

    // Loop-invariant B-operand coefficients: srB = u[k] * Pr, siB = u[k] * Pi
    // with P = a^{-k} * (Br + i Bi) for this lane's fixed k = 4q + 2*hh + r.
    v2f Pr[4], Pi[4];
    #pragma unroll
    for (int q = 0; q < 4; ++q) {
        #pragma unroll
        for (int r = 0; r < 2; ++r) {
            const int   k  = 4 * q + 2 * hh + r;
            const float ek = __expf(-lm * (float)k);
            float sn, cs; __sincosf(ph * (float)k, &sn, &cs);
            const float wR =  ek * cs, wI = -ek * sn;      // a^{-k}
            const float pr = wR * Br - wI * Bi;
            const float pi = wR * Bi + wI * Br;
            if (r == 0) { Pr[q].x = pr; Pi[q].x = pi; }
            else        { Pr[q].y = pr; Pi[q].y = pi; }
        }
    }

    // Loop-invariant output coefficients for this lane's taus (tau = r + 8*hh):
    //   val = CA*hr0 + CB*hi0 + QR*cs_r + QI*cs_i
    // where C, a^tau, a^{tau+1} are all folded in.
    float CA[8], CB[8], QR[8], QI[8];
    #pragma unroll
    for (int r = 0; r < 8; ++r) {
        const int   tau = r + 8 * hh;
        const float et  = __expf(lm * (float)tau);
        float sn, cs; __sincosf(ph * (float)tau, &sn, &cs);
        const float aR  = et * cs, aI = et * sn;           // a^tau
        const float aR1 = aR * a1r - aI * a1i;             // a^{tau+1}
        const float aI1 = aR * a1i + aI * a1r;
        CA[r] =  Cr * aR1 - Ci * aI1;
        CB[r] = -(Cr * aI1 + Ci * aR1);
        QR[r] =  Cr * aR  - Ci * aI;
        QI[r] = -(Cr * aI  + Ci * aR);
    }

    // Carry coefficients (tau = 15): H' = a^16 * H + a^15 * cs[15]
    const float e15 = __expf(lm * 15.0f);
    float s15, c15; __sincosf(ph * 15.0f, &s15, &c15);
    const float aR15 = e15 * c15, aI15 = e15 * s15;
    const float aR16 = aR15 * a1r - aI15 * a1i;
    const float aI16 = aR15 * a1i + aI15 * a1r;

    if (tid < S_DIM) { Hr[tid] = 0.0f; Hi[tid] = 0.0f; }
    __syncthreads();

    const size_t xbase = ((size_t)b * T_LEN) * D_DIM + (size_t)d;
    // Wave-relative LDS byte offset of this lane's xbuf slot (addr[31:0] of
    // the generic LDS pointer is the LDS offset per the flat-aperture rule).
    const unsigned xbuf_lds = (unsigned)(size_t)(&xbuf[tid & (CH - 1)]);

    for (int t0 = 0; t0 < T_LEN; t0 += CH) {
        if (tid < CH) {
            // Async global -> LDS copy of this chunk's 16 x values (ASYNCcnt).
            const float* gp = &x[xbase + (size_t)(t0 + tid) * D_DIM];
            asm volatile("global_load_async_to_lds_b32 %0, %1, off"
                         :: "v"(xbuf_lds), "v"(gp) : "memory");
            if (t0 + CH < T_LEN)
                __builtin_prefetch(&x[xbase + (size_t)(t0 + CH + tid) * D_DIM], 0, 3);
        }
        // Issuing wave drains its async ops; other waves have ASYNCcnt==0.
        asm volatile("s_wait_asynccnt 0x0" ::: "memory");
        __syncthreads();

        const float hr0 = Hr[s];
        const float hi0 = Hi[s];

        v8f csr = {}; v8f csi = {};
        #pragma unroll
        for (int q = 0; q < 4; ++q) {
            const int   k0 = 4 * q + 2 * hh;
            const float u0 = xbuf[k0];
            const float u1 = xbuf[k0 + 1];
            v2f srB = { u0 * Pr[q].x, u1 * Pr[q].y };
            v2f siB = { u0 * Pi[q].x, u1 * Pi[q].y };
            csr = __builtin_amdgcn_wmma_f32_16x16x4_f32(
                      false, Amat[q], false, srB, (short)0, csr, false, false);
            csi = __builtin_amdgcn_wmma_f32_16x16x4_f32(
                      false, Amat[q], false, siB, (short)0, csi, false, false);
        }

        // C/D layout: lane holds tau = r + 8*hh (reg r), column s = sl.
        #pragma unroll
        for (int r = 0; r < 8; ++r) {
            float val = CA[r] * hr0 + CB[r] * hi0 + QR[r] * csr[r] + QI[r] * csi[r];
            val = dpp_row_add<0x111>(val);   // row_shr:1
            val = dpp_row_add<0x112>(val);   // row_shr:2
            val = dpp_row_add<0x114>(val);   // row_shr:4
            val = dpp_row_add<0x118>(val);   // row_shr:8
            if (sl == 15) ybuf[wave][r + 8 * hh] = val;  // lanes 15 & 31
        }

        if (hh == 1) {   // carry H' = h[15]; same-wave LDS ordering is safe
            const float cr15 = csr[7], ci15 = csi[7];
            Hr[s] = aR16 * hr0 - aI16 * hi0 + aR15 * cr15 - aI15 * ci15;
            Hi[s] = aR16 * hi0 + aI16 * hr0 + aR15 * ci15 + aI15 * cr15;
        }
        __syncthreads();

        if (tid < CH) {
            const float yv = ybuf[0][tid] + ybuf[1][tid] +
                             ybuf[2][tid] + ybuf[3][tid] + Dd * xbuf[tid];
            y[xbase + (size_t)(t0 + tid) * D_DIM] = yv;
        }
        // top-of-loop barrier orders ybuf/xbuf reuse for next chunk
    }
}

extern "C" void kernel_launch(void* const* d_in, const int* in_sizes, int n_in,
                              void* d_out, int out_size, void* d_ws, size_t ws_size,
                              hipStream_t stream) {
    const float* x    = (const float*)d_in[0];
    const float* lam  = (const float*)d_in[1];
    const float* aph  = (const float*)d_in[2];
    const float* Bre  = (const float*)d_in[3];
    const float* Bim  = (const float*)d_in[4];
    const float* Cre  = (const float*)d_in[5];
    const float* Cim  = (const float*)d_in[6];
    const float* Ddia = (const float*)d_in[7];
    float* y = (float*)d_out;

    dim3 grid(BSZ * D_DIM);   // 256 workgroups, one per (b, d)
    dim3 block(128);          // 4 waves (wave32)
    lru_ssd_wmma_kernel<<<grid, block, 0, stream>>>(
        x, lam, aph, Bre, Bim, Cre, Cim, Ddia, y);
}